// Quantizer_1958505086982
// MI455X (gfx1250) — compile-verified
//
#include <hip/hip_runtime.h>
#include <hip/hip_bf16.h>
#include <math.h>

typedef float v2f __attribute__((ext_vector_type(2)));
typedef float v8f __attribute__((ext_vector_type(8)));

#define KCODES 512
#define DDIM   64
#define HW     4096
#define NPOS   131072            // 32*64*64
#define ZELEMS 8388608           // NPOS * DDIM

// d_out float offsets (tuple concatenated flat, in return order)
#define OFF_LOSS 0
#define OFF_ZQ   1UL
#define OFF_PERP 8388609UL
#define OFF_ENC  8388610UL
#define OFF_IDX  75497474UL

// ---------------------------------------------------------------------------
// Kernel 1: fused distance GEMM (V_WMMA_F32_16X16X4_F32) + argmin over codes.
// Block = 256 thr = 8 waves; each wave owns 16 rows of z.
// Embedding staged in LDS as K-pair-interleaved float2:
//   eP[pair*512 + code] = (e[code][2*pair], e[code][2*pair+1])
// so each WMMA B-operand is ONE contiguous ds_load_b64, conflict-free.
// ---------------------------------------------------------------------------
__global__ void __launch_bounds__(256)
vq_argmin_wmma(const float* __restrict__ z, const float* __restrict__ emb,
               int* __restrict__ idx_out)
{
    extern __shared__ float smem[];                 // eP[32*512]*2 + enorm[512]
    v2f*   eP    = (v2f*)smem;                      // [pair][code]
    float* enorm = smem + 2 * 32 * KCODES;

    const int tid = threadIdx.x;

    // cooperative fill: one float2 per element pair, code-major
    // global read 8B-aligned; LDS b64 writes stride-8B -> conflict-free
    for (int i = tid; i < 32 * KCODES; i += 256) {
        int code = i & (KCODES - 1);
        int pair = i >> 9;
        const v2f v = *(const v2f*)(emb + code * DDIM + 2 * pair);
        eP[pair * KCODES + code] = v;
    }
    __syncthreads();
    for (int k = tid; k < KCODES; k += 256) {
        float s = 0.f;
        #pragma unroll 8
        for (int p = 0; p < 32; ++p) {
            v2f v = eP[p * KCODES + k];
            s += v.x * v.x + v.y * v.y;
        }
        enorm[k] = s;
    }
    __syncthreads();

    const int lane = tid & 31;
    const int wave = tid >> 5;
    const int half = lane >> 4;                 // 0: K pair {0,1}; 1: {2,3}
    const int l15  = lane & 15;

    const int n0 = blockIdx.x * 128 + wave * 16;           // first row of tile
    const int b  = n0 >> 12;
    const int hw = n0 & 4095;
    const float* zb = z + (size_t)b * (DDIM * HW) + hw;    // + d*4096 + m

    // A operand: 16x4 f32 layout — lanes 0-15 rows, halves split K
    v2f a[16];
    #pragma unroll
    for (int kb = 0; kb < 16; ++kb) {
        int kd = 4 * kb + 2 * half;
        a[kb].x = zb[(size_t)kd       * HW + l15];
        a[kb].y = zb[(size_t)(kd + 1) * HW + l15];
    }

    float minv[8];
    int   minc[8];
    #pragma unroll
    for (int v = 0; v < 8; ++v) { minv[v] = 3.4e38f; minc[v] = 0; }

    for (int t = 0; t < KCODES / 16; ++t) {
        const int code = t * 16 + l15;          // this lane's code column
        v8f c = {0.f, 0.f, 0.f, 0.f, 0.f, 0.f, 0.f, 0.f};
        #pragma unroll
        for (int kb = 0; kb < 16; ++kb) {
            // B(kd..kd+1, code) = e[code][kd..kd+1] = eP[(2*kb+half)][code]
            v2f bb = eP[(2 * kb + half) * KCODES + code];
            c = __builtin_amdgcn_wmma_f32_16x16x4_f32(
                    false, a[kb], false, bb, (short)0, c, false, false);
        }
        float en = enorm[code];
        #pragma unroll
        for (int v = 0; v < 8; ++v) {           // row = v + 8*half, col = code
            float val = en - 2.0f * c[v];       // ||e||^2 - 2 z.e  (||z||^2 const)
            if (val < minv[v]) { minv[v] = val; minc[v] = code; }
        }
    }

    // min-reduce across the 16 lanes of each half (first-index tie-break)
    #pragma unroll
    for (int m = 1; m <= 8; m <<= 1) {
        #pragma unroll
        for (int v = 0; v < 8; ++v) {
            float ov = __shfl_xor(minv[v], m, 32);
            int   oc = __shfl_xor(minc[v], m, 32);
            if (ov < minv[v] || (ov == minv[v] && oc < minc[v])) {
                minv[v] = ov; minc[v] = oc;
            }
        }
    }
    if (l15 == 0) {
        #pragma unroll
        for (int v = 0; v < 8; ++v)
            idx_out[n0 + half * 8 + v] = minc[v];
    }
}

// ---------------------------------------------------------------------------
// Kernel 2: zero the one-hot region (268 MB — the bandwidth-dominant stream)
// ---------------------------------------------------------------------------
__global__ void zero_enc(float2* __restrict__ enc, long n2)
{
    long i = (long)blockIdx.x * blockDim.x + threadIdx.x;
    long stride = (long)gridDim.x * blockDim.x;
    float2 zz; zz.x = 0.f; zz.y = 0.f;
    for (; i < n2; i += stride) enc[i] = zz;
}

// ---------------------------------------------------------------------------
// Kernel 3: scatter ones + integer histogram (deterministic)
// ---------------------------------------------------------------------------
__global__ void scatter_onehot(const int* __restrict__ idx,
                               float* __restrict__ enc, int* __restrict__ hist)
{
    int n = blockIdx.x * blockDim.x + threadIdx.x;
    if (n < NPOS) {
        int k = idx[n];
        enc[(size_t)n * KCODES + k] = 1.0f;
        atomicAdd(&hist[k], 1);
    }
}

// ---------------------------------------------------------------------------
// Kernel 4: z_q gather written in [B,C,H,W] order (coalesced over hw)
// ---------------------------------------------------------------------------
__global__ void gather_zq(const int* __restrict__ idx,
                          const float* __restrict__ emb, float* __restrict__ zq)
{
    size_t i = (size_t)blockIdx.x * blockDim.x + threadIdx.x;
    if (i < ZELEMS) {
        int hw = (int)(i & 4095);
        int c  = (int)((i >> 12) & 63);
        int b  = (int)(i >> 18);
        int n  = (b << 12) | hw;
        zq[i] = emb[idx[n] * DDIM + c];
    }
}

// ---------------------------------------------------------------------------
// Kernel 5: loss partials  sum((z_q - z)^2), deterministic tree reduce
// ---------------------------------------------------------------------------
__global__ void __launch_bounds__(256)
loss_partial(const float* __restrict__ z, const float* __restrict__ emb,
             const int* __restrict__ idx, float* __restrict__ partial)
{
    __shared__ float red[256];
    float s = 0.f;
    for (size_t i = (size_t)blockIdx.x * 256 + threadIdx.x; i < ZELEMS;
         i += (size_t)256 * 1024) {
        int hw = (int)(i & 4095);
        int c  = (int)((i >> 12) & 63);
        int b  = (int)(i >> 18);
        int n  = (b << 12) | hw;
        float d = emb[idx[n] * DDIM + c] - z[i];
        s += d * d;
    }
    red[threadIdx.x] = s;
    __syncthreads();
    for (int off = 128; off > 0; off >>= 1) {
        if (threadIdx.x < off) red[threadIdx.x] += red[threadIdx.x + off];
        __syncthreads();
    }
    if (threadIdx.x == 0) partial[blockIdx.x] = red[0];
}

// ---------------------------------------------------------------------------
// Kernel 6: finalize loss + perplexity (faithful to the reference's formula)
// ---------------------------------------------------------------------------
__global__ void __launch_bounds__(256)
finalize(const float* __restrict__ partial, const int* __restrict__ hist,
         float* __restrict__ loss_out, float* __restrict__ perp_out)
{
    __shared__ float red[256];
    float s = 0.f;
    for (int i = threadIdx.x; i < 1024; i += 256) s += partial[i];
    red[threadIdx.x] = s;
    __syncthreads();
    for (int off = 128; off > 0; off >>= 1) {
        if (threadIdx.x < off) red[threadIdx.x] += red[threadIdx.x + off];
        __syncthreads();
    }
    if (threadIdx.x == 0)
        *loss_out = 1.25f * red[0] / (float)ZELEMS;   // (1 + BETA) * MSE
    __syncthreads();

    float p = 0.f;
    for (int k = threadIdx.x; k < KCODES; k += 256) {
        float pk = (float)hist[k] / (float)NPOS;
        p += pk + logf(pk + 1e-10f);
    }
    red[threadIdx.x] = p;
    __syncthreads();
    for (int off = 128; off > 0; off >>= 1) {
        if (threadIdx.x < off) red[threadIdx.x] += red[threadIdx.x + off];
        __syncthreads();
    }
    if (threadIdx.x == 0) *perp_out = expf(-red[0]);
}

// ---------------------------------------------------------------------------
extern "C" void kernel_launch(void* const* d_in, const int* in_sizes, int n_in,
                              void* d_out, int out_size, void* d_ws, size_t ws_size,
                              hipStream_t stream)
{
    const float* z   = (const float*)d_in[0];   // [32, 64, 64, 64]
    const float* emb = (const float*)d_in[1];   // [512, 64]

    float* out   = (float*)d_out;
    float* loss  = out + OFF_LOSS;
    float* zq    = out + OFF_ZQ;
    float* perp  = out + OFF_PERP;
    float* enc   = out + OFF_ENC;
    int*   idx   = (int*)(out + OFF_IDX);       // int32 bits in indices slot

    int*   hist    = (int*)d_ws;                // [512]
    float* partial = (float*)d_ws + KCODES;     // [1024]

    hipMemsetAsync(hist, 0, KCODES * sizeof(int), stream);

    // 1) WMMA distance GEMM + argmin  (dyn LDS: 32*512 float2 + 512 f32)
    const size_t lds_bytes = (size_t)(2 * 32 * KCODES + KCODES) * sizeof(float);
    vq_argmin_wmma<<<NPOS / 128, 256, lds_bytes, stream>>>(z, emb, idx);

    // 2) zero one-hot region (float2-aligned: byte off 33554440 % 8 == 0)
    zero_enc<<<8192, 256, 0, stream>>>((float2*)enc,
                                       (long)NPOS * KCODES / 2);

    // 3) scatter ones + histogram
    scatter_onehot<<<NPOS / 256, 256, 0, stream>>>(idx, enc, hist);

    // 4) z_q gather in [B,C,H,W] order
    gather_zq<<<ZELEMS / 256, 256, 0, stream>>>(idx, emb, zq);

    // 5) loss partials + 6) finalize
    loss_partial<<<1024, 256, 0, stream>>>(z, emb, idx, partial);
    finalize<<<1, 256, 0, stream>>>(partial, hist, loss, perp);
}